// InputLayer_27865747816678
// MI455X (gfx1250) — compile-verified
//
#include <hip/hip_runtime.h>
#include <hip/hip_bf16.h>
#include <math.h>

// Problem constants (fixed by setup_inputs)
#define NN   4096      // N
#define QQ   4         // Q
#define LT   8         // Lt
#define MM   128       // M (inducing)
#define DD   32        // D = Lt*Q
#define SIG2 0.001f
#define JIT  1e-5f
#define NSLICE 8       // n-dimension split of psi2 reduction

typedef __attribute__((ext_vector_type(2))) float v2f;
typedef __attribute__((ext_vector_type(8))) float v8f;

// D(16x16,f32) = A(16x4,f32) * B(4x16,f32) + C   — CDNA5 V_WMMA_F32_16X16X4_F32
__device__ __forceinline__ v8f wmma4(v2f a, v2f b, v8f c) {
  return __builtin_amdgcn_wmma_f32_16x16x4_f32(false, a, false, b, (short)0, c, false, false);
}

// CDNA5 async global->LDS copy (ASYNCcnt-tracked, bypasses VGPRs).
// VDST = per-lane LDS byte address, VADDR = per-lane 64-bit global address.
__device__ __forceinline__ void async_ld128(void* lds_ptr, const void* gptr) {
  unsigned lo = (unsigned)(uintptr_t)lds_ptr;  // low 32 bits of generic ptr == LDS offset
  unsigned long long ga = (unsigned long long)(uintptr_t)gptr;
  asm volatile("global_load_async_to_lds_b128 %0, %1, off" :: "v"(lo), "v"(ga) : "memory");
}
__device__ __forceinline__ void wait_async0() {
  asm volatile("s_wait_asynccnt 0" ::: "memory");
}

// ---- workspace layout (float offsets) ----
#define OFF_R1    0          // N*D  Xmu/denom
#define OFF_R2    131072     // N*D  1/denom
#define OFF_AM    262144     // N*D  a*Xmu
#define OFF_AA    393216     // N*D  a
#define OFF_C1    524288     // N    log(var) - 0.5*(sum log1p + s1)
#define OFF_G     528384     // N    var^2 exp(-0.5 sum log1p(2v/l2) - T1)
#define OFF_EN    532480     // N*M  E[n,m] = exp(T2 - 0.25 T3)
#define OFF_PSI1  1056768    // N*M
#define OFF_PSI2  1581056    // M*M
#define OFF_EDZ   1597440    // M*M  exp(-dZ2)
#define OFF_KUU   1613824    // M*M  (Cholesky'd in place)
#define OFF_TMPM  1630208    // M*M  L^-1 psi2
#define OFF_AAT   1646592    // M*M
#define OFF_BM    1662976    // M*M  scratch solve / B=AAT+I (chol'd)
#define OFF_P1X   1679360    // M*Q  psi1^T X_mo
#define OFF_WC    1679872    // M*Q*2  W then c
#define OFF_P2P   1680896    // NSLICE*64*256 psi2 slice partials

// ---------------- K0: per-n hankel-derived precompute ----------------
__global__ void k0_pre(const float* __restrict__ Xm, const float* __restrict__ Xv,
                       const float* __restrict__ ls, const float* __restrict__ varp,
                       float* __restrict__ r1, float* __restrict__ r2,
                       float* __restrict__ am, float* __restrict__ aa,
                       float* __restrict__ c1, float* __restrict__ g) {
  int n = blockIdx.x * blockDim.x + threadIdx.x;
  if (n >= NN) return;
  float var = varp[0];
  float sld = 0.f, s1 = 0.f, sld2 = 0.f, T1 = 0.f;
#pragma unroll
  for (int d = 0; d < DD; ++d) {
    int l = d >> 2, q = d & 3;
    float xm = Xm[(n + l) * QQ + q];
    float xv = Xv[(n + l) * QQ + q];
    float L2 = ls[d] * ls[d];
    float inv = 1.0f / (L2 + xv);
    float r1v = xm * inv;
    sld  += log1pf(xv / L2);
    s1   += xm * r1v;
    float av = 1.0f / (2.0f * xv + L2);
    sld2 += log1pf(2.0f * xv / L2);
    T1   += av * xm * xm;
    r1[n * DD + d] = r1v;
    r2[n * DD + d] = inv;
    aa[n * DD + d] = av;
    am[n * DD + d] = av * xm;
  }
  c1[n] = __logf(var) - 0.5f * (sld + s1);
  g[n]  = var * var * __expf(-0.5f * sld2 - T1);
}

// ---------------- K1: Kuu and exp(-dZ2) ----------------
__global__ void k1_kuu(const float* __restrict__ Zp, const float* __restrict__ ls,
                       const float* __restrict__ varp,
                       float* __restrict__ Kuu, float* __restrict__ Edz) {
  int idx = blockIdx.x * blockDim.x + threadIdx.x;
  if (idx >= MM * MM) return;
  int m = idx >> 7, k = idx & (MM - 1);
  float sK = 0.f, sE = 0.f;
#pragma unroll
  for (int d = 0; d < DD; ++d) {
    float df = Zp[m * DD + d] - Zp[k * DD + d];
    float L2 = ls[d] * ls[d];
    float d2 = df * df;
    sK += d2 / L2;
    sE += d2 / (4.0f * L2);
  }
  Kuu[idx] = varp[0] * __expf(-0.5f * sK) + ((m == k) ? JIT : 0.0f);
  Edz[idx] = __expf(-sE);
}

// ---------------- K2: fused WMMA GEMMs -> E[n,m] and psi1[n,m] ----------------
// Two N x M contractions over K=64 ([am|aa]x[Z|-0.25 Z^2], [r1|r2]x[Z|-0.5 Z^2]).
// Block: 128 threads (4 waves), each block handles 64 n-rows, loops all 8 m-tiles.
__global__ void k2_gemm(const float* __restrict__ Zp,
                        const float* __restrict__ r1, const float* __restrict__ r2,
                        const float* __restrict__ am, const float* __restrict__ aa,
                        const float* __restrict__ c1,
                        float* __restrict__ En, float* __restrict__ psi1) {
  __shared__ float Au[64 * 64];   // [row][k] : am | aa
  __shared__ float Ap[64 * 64];   // [row][k] : r1 | r2
  __shared__ float c1s[64];
  int tid = threadIdx.x;
  int n0g = blockIdx.x * 64;
  for (int i = tid; i < 64 * 64; i += 128) {
    int r = i >> 6, c = i & 63;
    int n = n0g + r;
    Au[i] = (c < DD) ? am[n * DD + c] : aa[n * DD + (c - DD)];
    Ap[i] = (c < DD) ? r1[n * DD + c] : r2[n * DD + (c - DD)];
  }
  for (int i = tid; i < 64; i += 128) c1s[i] = c1[n0g + i];
  __syncthreads();

  int wave = tid >> 5, lane = tid & 31, hi = lane >> 4, lr = lane & 15;
  int rbase = wave * 16;
  for (int ct = 0; ct < 8; ++ct) {
    int mcol = ct * 16 + lr;
    v8f accU = {};
    v8f accP = {};
#pragma unroll
    for (int s = 0; s < 16; ++s) {
      int kk = 4 * s + 2 * hi;
      v2f aU; aU.x = Au[(rbase + lr) * 64 + kk]; aU.y = Au[(rbase + lr) * 64 + kk + 1];
      v2f aP; aP.x = Ap[(rbase + lr) * 64 + kk]; aP.y = Ap[(rbase + lr) * 64 + kk + 1];
      v2f bU, bP;
      if (kk < DD) {           // Z part (kk and kk+1 both < 32 since s < 8)
        float z0 = Zp[mcol * DD + kk];
        float z1 = Zp[mcol * DD + kk + 1];
        bU.x = z0; bU.y = z1;
        bP.x = z0; bP.y = z1;
      } else {                 // Z^2 part
        float z0 = Zp[mcol * DD + kk - DD];
        float z1 = Zp[mcol * DD + kk - DD + 1];
        bU.x = -0.25f * z0 * z0; bU.y = -0.25f * z1 * z1;
        bP.x = -0.5f  * z0 * z0; bP.y = -0.5f  * z1 * z1;
      }
      accU = wmma4(aU, bU, accU);
      accP = wmma4(aP, bP, accP);
    }
#pragma unroll
    for (int v = 0; v < 8; ++v) {
      int rl = rbase + v + 8 * hi;
      int n  = n0g + rl;
      En[n * MM + mcol]   = __expf(accU[v]);
      psi1[n * MM + mcol] = __expf(c1s[rl] + accP[v]);
    }
  }
}

// ---------------- K3: psi2 slice partials (the heavy one) ----------------
// part[slice,tile,mi,ki] = sum_{n in slice} g[n] E[n,m] E[n,k] exp(-0.5 sum_q a[n,q] Zm Zk)
// Grid: 512 blocks = 64 tiles x 8 n-slices. Block: 128 threads (4 waves).
// Staging double-buffered via CDNA5 GLOBAL_LOAD_ASYNC_TO_LDS_B128 (ASYNCcnt).
__global__ void k3_psi2(const float* __restrict__ Zp, const float* __restrict__ aaw,
                        const float* __restrict__ gw, const float* __restrict__ En,
                        float* __restrict__ part) {
  __shared__ __align__(16) float Zm[16 * DD];
  __shared__ __align__(16) float Zk[16 * DD];
  __shared__ __align__(16) float Ach[2][64 * DD];
  __shared__ __align__(16) float Em[2][64 * 16];
  __shared__ __align__(16) float Ek[2][64 * 16];
  __shared__ __align__(16) float gs[2][64];
  __shared__ float red[4 * 16 * 32];
  int tileId = blockIdx.x & 63;
  int slice  = blockIdx.x >> 6;
  int mt = tileId >> 3, kt = tileId & 7;
  int tid = threadIdx.x;
  for (int i = tid; i < 16 * DD; i += 128) {
    Zm[i] = Zp[(mt * 16) * DD + i];
    Zk[i] = Zp[(kt * 16) * DD + i];
  }
  int wave = tid >> 5, lane = tid & 31, hi = lane >> 4, lr = lane & 15;

  const int CHUNKS = (NN / NSLICE) / 64;   // 8 chunks of 64 rows per slice
  const int nsb = slice * (NN / NSLICE);

  // issue async stage of chunk 0 into buffer 0
  {
    int nb = nsb;
    for (int i = tid * 4; i < 64 * DD; i += 512)
      async_ld128(&Ach[0][i], &aaw[nb * DD + i]);
    for (int i = tid * 4; i < 64 * 16; i += 512) {
      int r = i >> 4, j = i & 15;
      async_ld128(&Em[0][i], &En[(nb + r) * MM + mt * 16 + j]);
      async_ld128(&Ek[0][i], &En[(nb + r) * MM + kt * 16 + j]);
    }
    if (tid < 16) async_ld128(&gs[0][tid * 4], &gw[nb + tid * 4]);
  }
  wait_async0();
  __syncthreads();

  v2f zkf[8];                 // B-side Zk fragments, constant over n
#pragma unroll
  for (int s = 0; s < 8; ++s) {
    int kk = 4 * s + 2 * hi;
    zkf[s].x = Zk[lr * DD + kk];
    zkf[s].y = Zk[lr * DD + kk + 1];
  }
  float accv[16];
#pragma unroll
  for (int i = 0; i < 16; ++i) accv[i] = 0.f;

  for (int nc = 0; nc < CHUNKS; ++nc) {
    int cur = nc & 1;
    // prefetch next chunk into the other buffer while we compute
    if (nc + 1 < CHUNKS) {
      int nb = nsb + (nc + 1) * 64;
      int nxt = 1 - cur;
      for (int i = tid * 4; i < 64 * DD; i += 512)
        async_ld128(&Ach[nxt][i], &aaw[nb * DD + i]);
      for (int i = tid * 4; i < 64 * 16; i += 512) {
        int r = i >> 4, j = i & 15;
        async_ld128(&Em[nxt][i], &En[(nb + r) * MM + mt * 16 + j]);
        async_ld128(&Ek[nxt][i], &En[(nb + r) * MM + kt * 16 + j]);
      }
      if (tid < 16) async_ld128(&gs[nxt][tid * 4], &gw[nb + tid * 4]);
    }

    v2f af[8];
    int arow = wave * 16 + lr;
#pragma unroll
    for (int s = 0; s < 8; ++s) {
      int kk = 4 * s + 2 * hi;
      af[s].x = Ach[cur][arow * DD + kk];
      af[s].y = Ach[cur][arow * DD + kk + 1];
    }
    float gek[8];
#pragma unroll
    for (int v = 0; v < 8; ++v) {
      int rl = wave * 16 + v + 8 * hi;
      gek[v] = gs[cur][rl] * Ek[cur][rl * 16 + lr];
    }
    for (int mi = 0; mi < 16; ++mi) {
      v8f acc = {};
#pragma unroll
      for (int s = 0; s < 8; ++s) {
        int kk = 4 * s + 2 * hi;
        v2f b;
        b.x = Zm[mi * DD + kk]     * zkf[s].x;
        b.y = Zm[mi * DD + kk + 1] * zkf[s].y;
        acc = wmma4(af[s], b, acc);
      }
      float ssum = 0.f;
#pragma unroll
      for (int v = 0; v < 8; ++v) {
        int rl = wave * 16 + v + 8 * hi;
        ssum += gek[v] * Em[cur][rl * 16 + mi] * __expf(-0.5f * acc[v]);
      }
      accv[mi] += ssum;
    }
    wait_async0();          // next buffer landed
    __syncthreads();        // across all waves before swap
  }
  // deterministic cross-wave / cross-half reduction via LDS
#pragma unroll
  for (int mi = 0; mi < 16; ++mi) red[(wave * 16 + mi) * 32 + lane] = accv[mi];
  __syncthreads();
  for (int p = tid; p < 256; p += 128) {
    int mi = p >> 4, ki = p & 15;
    float s = 0.f;
    for (int w = 0; w < 4; ++w)
      s += red[(w * 16 + mi) * 32 + ki] + red[(w * 16 + mi) * 32 + ki + 16];
    part[(slice * 64 + tileId) * 256 + p] = s;
  }
}

// ---------------- K3b: deterministic slice reduction + Edz scaling ----------------
__global__ void k3b_reduce(const float* __restrict__ part, const float* __restrict__ Edz,
                           float* __restrict__ psi2) {
  int p = blockIdx.x * blockDim.x + threadIdx.x;
  if (p >= 64 * 256) return;
  int tile = p >> 8, e = p & 255;
  int mi = e >> 4, ki = e & 15;
  int gm = (tile >> 3) * 16 + mi, gk = (tile & 7) * 16 + ki;
  float s = 0.f;
#pragma unroll
  for (int sl = 0; sl < NSLICE; ++sl) s += part[(sl * 64 + tile) * 256 + e];
  psi2[gm * MM + gk] = Edz[gm * MM + gk] * s;
}

// ---------------- K4: reductions + Cholesky + solves + bound ----------------
__device__ void chol128(float* A, int tid) {
  for (int j = 0; j < MM; ++j) {
    if (tid == 0) A[j * MM + j] = sqrtf(A[j * MM + j]);
    __syncthreads();
    float dinv = 1.0f / A[j * MM + j];
    for (int i = j + 1 + tid; i < MM; i += 256) A[i * MM + j] *= dinv;
    __syncthreads();
    for (int i = j + 1; i < MM; ++i) {
      float lij = A[i * MM + j];
      for (int k = j + 1 + tid; k <= i; k += 256) A[i * MM + k] -= lij * A[k * MM + j];
    }
    __syncthreads();
  }
}

__global__ void k4_final(const float* __restrict__ Xm, const float* __restrict__ Xv,
                         const float* __restrict__ varp,
                         const float* __restrict__ psi1, const float* __restrict__ psi2,
                         float* __restrict__ Kuu, float* __restrict__ tmpM,
                         float* __restrict__ AAT, float* __restrict__ Bm,
                         float* __restrict__ P1X, float* __restrict__ Wc,
                         float* __restrict__ out) {
  __shared__ double dred[256];
  __shared__ float sh[8];
  int tid = threadIdx.x;

  // --- scalar sums over X slices ---
  double a0 = 0, a1 = 0, a2 = 0, a3 = 0;
  for (int i = tid; i < NN * QQ; i += 256) {
    float xm = Xm[LT * QQ + i];
    float xv = Xv[LT * QQ + i];
    a0 += (double)xv;
    a1 += (double)xm * xm;
    a2 += (double)__logf(xv);
  }
  for (int i = tid; i < LT * QQ; i += 256) {
    float xm = Xm[i], xv = Xv[i];
    a3 += (double)xm * xm + (double)xv;
  }
  double vals[4] = {a0, a1, a2, a3};
  for (int r = 0; r < 4; ++r) {
    dred[tid] = vals[r]; __syncthreads();
    for (int s = 128; s > 0; s >>= 1) { if (tid < s) dred[tid] += dred[tid + s]; __syncthreads(); }
    if (tid == 0) sh[r] = (float)dred[0];
    __syncthreads();
  }

  // --- P1X = psi1^T @ X_mo  (M x Q) ---
  for (int p = tid; p < MM * QQ; p += 256) {
    int m = p >> 2, q = p & 3;
    float s = 0.f;
    for (int n = 0; n < NN; ++n) s += psi1[n * MM + m] * Xm[(n + LT) * QQ + q];
    P1X[p] = s;
  }
  __syncthreads();

  // --- L = chol(Kuu) (in place, lower) ---
  chol128(Kuu, tid);

  // --- tmpM = L^-1 psi2 (column-parallel forward solve) ---
  if (tid < MM) {
    int c = tid;
    for (int j = 0; j < MM; ++j) {
      float s = psi2[j * MM + c];
      for (int k = 0; k < j; ++k) s -= Kuu[j * MM + k] * tmpM[k * MM + c];
      tmpM[j * MM + c] = s / Kuu[j * MM + j];
    }
  }
  __syncthreads();

  // --- y = L^-1 tmpM^T (into Bm), AAT = y / sigma2 ---
  if (tid < MM) {
    int c = tid;
    for (int j = 0; j < MM; ++j) {
      float s = tmpM[c * MM + j];
      for (int k = 0; k < j; ++k) s -= Kuu[j * MM + k] * Bm[k * MM + c];
      Bm[j * MM + c] = s / Kuu[j * MM + j];
    }
  }
  __syncthreads();
  for (int i = tid; i < MM * MM; i += 256) AAT[i] = Bm[i] * (1.0f / SIG2);
  __syncthreads();

  // --- trace(AAT) ---
  double tr = 0;
  for (int i = tid; i < MM; i += 256) tr += (double)AAT[i * MM + i];
  dred[tid] = tr; __syncthreads();
  for (int s = 128; s > 0; s >>= 1) { if (tid < s) dred[tid] += dred[tid + s]; __syncthreads(); }
  if (tid == 0) sh[4] = (float)dred[0];
  __syncthreads();

  // --- B = AAT + I, LB = chol(B), log_det_B ---
  for (int i = tid; i < MM * MM; i += 256) {
    int r = i >> 7, c = i & (MM - 1);
    Bm[i] = AAT[i] + ((r == c) ? 1.0f : 0.0f);
  }
  __syncthreads();
  chol128(Bm, tid);
  double ld = 0;
  for (int i = tid; i < MM; i += 256) ld += 2.0 * (double)__logf(Bm[i * MM + i]);
  dred[tid] = ld; __syncthreads();
  for (int s = 128; s > 0; s >>= 1) { if (tid < s) dred[tid] += dred[tid + s]; __syncthreads(); }
  if (tid == 0) sh[5] = (float)dred[0];
  __syncthreads();

  // --- W = L^-1 P1X ; cvec = LB^-1 W ; c = cvec / sigma2 ---
  float* W = Wc;
  float* cv = Wc + MM * QQ;
  if (tid < QQ) {
    int q = tid;
    for (int j = 0; j < MM; ++j) {
      float s = P1X[j * QQ + q];
      for (int k = 0; k < j; ++k) s -= Kuu[j * MM + k] * W[k * QQ + q];
      W[j * QQ + q] = s / Kuu[j * MM + j];
    }
    for (int j = 0; j < MM; ++j) {
      float s = W[j * QQ + q];
      for (int k = 0; k < j; ++k) s -= Bm[j * MM + k] * cv[k * QQ + q];
      cv[j * QQ + q] = s / Bm[j * MM + j];
    }
  }
  __syncthreads();
  double sc2 = 0;
  for (int i = tid; i < MM * QQ; i += 256) {
    float c = cv[i] * (1.0f / SIG2);
    sc2 += (double)c * c;
  }
  dred[tid] = sc2; __syncthreads();
  for (int s = 128; s > 0; s >>= 1) { if (tid < s) dred[tid] += dred[tid + s]; __syncthreads(); }

  if (tid == 0) {
    float sumXvo = sh[0], sumXmo2 = sh[1], sumLogXvo = sh[2], sumMb = sh[3];
    float trAAT = sh[4], logDetB = sh[5];
    float sumc2 = (float)dred[0];
    float var = varp[0];
    const float two_pi = 6.2831853071795864769f;
    float NQ = (float)(NN * QQ);
    float psi0 = (float)NN * var;
    float bound = -0.5f * NQ * __logf(two_pi * SIG2);
    bound += -0.5f / SIG2 * (sumXvo + sumXmo2);
    bound += -0.5f * (float)QQ * (psi0 / SIG2 - trAAT);
    bound += -0.5f * (float)QQ * logDetB;
    bound += 0.5f * sumc2;
    bound += 0.5f * sumLogXvo + 0.5f * NQ * __logf(two_pi);
    bound += -(float)(LT * QQ) * __logf(two_pi) - 0.5f * sumMb;
    out[0] = bound;
  }
}

extern "C" void kernel_launch(void* const* d_in, const int* in_sizes, int n_in,
                              void* d_out, int out_size, void* d_ws, size_t ws_size,
                              hipStream_t stream) {
  (void)in_sizes; (void)n_in; (void)out_size; (void)ws_size;
  const float* Xmean = (const float*)d_in[0];
  const float* Xvar  = (const float*)d_in[1];
  const float* Zp    = (const float*)d_in[2];
  const float* ls    = (const float*)d_in[3];
  const float* varp  = (const float*)d_in[4];
  float* w = (float*)d_ws;

  k0_pre<<<NN / 256, 256, 0, stream>>>(Xmean, Xvar, ls, varp,
      w + OFF_R1, w + OFF_R2, w + OFF_AM, w + OFF_AA, w + OFF_C1, w + OFF_G);

  k1_kuu<<<(MM * MM) / 256, 256, 0, stream>>>(Zp, ls, varp, w + OFF_KUU, w + OFF_EDZ);

  k2_gemm<<<NN / 64, 128, 0, stream>>>(Zp,
      w + OFF_R1, w + OFF_R2, w + OFF_AM, w + OFF_AA, w + OFF_C1,
      w + OFF_EN, w + OFF_PSI1);

  k3_psi2<<<64 * NSLICE, 128, 0, stream>>>(Zp, w + OFF_AA, w + OFF_G, w + OFF_EN,
      w + OFF_P2P);

  k3b_reduce<<<(64 * 256) / 256, 256, 0, stream>>>(w + OFF_P2P, w + OFF_EDZ,
      w + OFF_PSI2);

  k4_final<<<1, 256, 0, stream>>>(Xmean, Xvar, varp,
      w + OFF_PSI1, w + OFF_PSI2, w + OFF_KUU, w + OFF_TMPM, w + OFF_AAT,
      w + OFF_BM, w + OFF_P1X, w + OFF_WC, (float*)d_out);
}